// RegionProposalNetwork_67233418051947
// MI455X (gfx1250) — compile-verified
//
#include <hip/hip_runtime.h>
#include <stdint.h>

#define PRE_NMS    2000u
#define POST_NMS   1000u
#define NMS_THR    0.7f
#define MIN_SIZE   1.0f
#define BBOX_CLIP  4.135166556742356f   // log(1000/16)
#define MASK_WORDS 63u                  // ceil(2000/32)
#define SORT_N     2048u
#define TILE       1024u                // 256 lanes x 4 floats per async tile

// ---------------- CDNA5 async global->LDS path (guarded) ----------------
#if defined(__HIP_DEVICE_COMPILE__) && defined(__gfx1250__) && \
    __has_builtin(__builtin_amdgcn_global_load_async_to_lds_b128) && \
    __has_builtin(__builtin_amdgcn_s_wait_asynccnt)
#define HAVE_ASYNC_LDS 1
typedef int async_v4i __attribute__((vector_size(16)));
#endif

// Monotonic float->uint key: order of keys == order of floats.
__device__ __forceinline__ unsigned fkey(float f) {
    unsigned u = __float_as_uint(f);
    return (u & 0x80000000u) ? ~u : (u | 0x80000000u);
}

// ---------------- Pass 0: zero a u32 range ----------------
__global__ void zero_u32_kernel(unsigned* __restrict__ p, unsigned n) {
    unsigned i = blockIdx.x * blockDim.x + threadIdx.x;
    if (i < n) p[i] = 0u;
}

// ---------------- Pass 1: histogram of key[31:16] ----------------
// Streams objectness through LDS with a double-buffered CDNA5 async pipeline:
// 2 tiles in flight, s_wait_asynccnt 1 in steady state (async loads complete
// in order), so HBM latency is hidden behind the previous tile's binning.
__global__ void hist_hi_kernel(const float* __restrict__ obj, unsigned N,
                               unsigned* __restrict__ hist) {
#ifdef HAVE_ASYNC_LDS
    __shared__ float tile[2][TILE];
    const unsigned lane4 = threadIdx.x * 4u;
    const unsigned step  = gridDim.x * TILE;
    unsigned next = blockIdx.x * TILE;
    int cnt = 0;
    if (next + TILE <= N) {
        __builtin_amdgcn_global_load_async_to_lds_b128(
            (async_v4i*)(obj + next + lane4), (async_v4i*)(&tile[0][lane4]), 0, 0);
        next += step; cnt = 1;
        if (next + TILE <= N) {
            __builtin_amdgcn_global_load_async_to_lds_b128(
                (async_v4i*)(obj + next + lane4), (async_v4i*)(&tile[1][lane4]), 0, 0);
            next += step; cnt = 2;
        }
    }
    int b = 0;
    while (cnt == 2) {
        __builtin_amdgcn_s_wait_asynccnt(1);   // oldest tile landed
#pragma unroll
        for (int q = 0; q < 4; ++q) {
            unsigned k = fkey(tile[b][lane4 + q]);
            atomicAdd(&hist[k >> 16], 1u);
        }
        if (next + TILE <= N) {                // refill freed buffer
            __builtin_amdgcn_global_load_async_to_lds_b128(
                (async_v4i*)(obj + next + lane4), (async_v4i*)(&tile[b][lane4]), 0, 0);
            next += step;
        } else {
            cnt = 1;
        }
        b ^= 1;
    }
    if (cnt == 1) {
        __builtin_amdgcn_s_wait_asynccnt(0);
#pragma unroll
        for (int q = 0; q < 4; ++q) {
            unsigned k = fkey(tile[b][lane4 + q]);
            atomicAdd(&hist[k >> 16], 1u);
        }
    }
    if (next < N) {                            // at most one partial tile
#pragma unroll
        for (unsigned q = 0; q < 4u; ++q) {
            unsigned i = next + lane4 + q;
            if (i < N) { unsigned k = fkey(obj[i]); atomicAdd(&hist[k >> 16], 1u); }
        }
    }
#else
    const unsigned lane4  = threadIdx.x * 4u;
    const unsigned stride = gridDim.x * blockDim.x * 4u;
    for (unsigned t0 = blockIdx.x * (blockDim.x * 4u); t0 < N; t0 += stride) {
        unsigned i = t0 + lane4;
        for (unsigned q = 0; q < 4u; ++q)
            if (i + q < N) { unsigned k = fkey(obj[i + q]); atomicAdd(&hist[k >> 16], 1u); }
    }
#endif
}

// ---------------- Pass 2/5: find threshold bin via suffix scan ----------------
// phase 0: target=PRE_NMS -> meta[0]=hiBin meta[1]=cntStrictAboveHi meta[2]=need
// phase 1: target=meta[2] -> meta[3]=thrKey meta[4]=strictGreater meta[5]=needEq
//                            meta[6]=meta[7]=0 (atomic counters)
__global__ void scan_hist_kernel(const unsigned* __restrict__ hist,
                                 unsigned* __restrict__ meta, int phase) {
    __shared__ unsigned csum[256];
    __shared__ unsigned pick[2];
    const int t = threadIdx.x;
    const unsigned target = (phase == 0) ? PRE_NMS : meta[2];
    unsigned s = 0;
    for (int b = 0; b < 256; ++b) s += hist[t * 256 + b];
    csum[t] = s;
    __syncthreads();
    if (t == 0) {
        unsigned cum = 0; int c = 255;
        for (; c > 0; --c) { if (cum + csum[c] >= target) break; cum += csum[c]; }
        pick[0] = (unsigned)c; pick[1] = cum;
    }
    __syncthreads();
    const int c = (int)pick[0];
    csum[t] = hist[c * 256 + t];
    __syncthreads();
    if (t == 0) {
        unsigned cum = pick[1]; int b = 255;
        for (; b > 0; --b) { if (cum + csum[b] >= target) break; cum += csum[b]; }
        unsigned bin = (unsigned)(c * 256 + b);
        if (phase == 0) {
            meta[0] = bin;
            meta[1] = cum;                 // strictly above hi bin
            meta[2] = target - cum;        // still needed inside hi bin
        } else {
            meta[3] = (meta[0] << 16) | bin;   // exact 32-bit threshold key
            meta[4] = meta[1] + cum;           // count strictly > thrKey
            meta[5] = target - cum;            // # equal-to-threshold to take
            meta[6] = 0u;                      // atomic ctr: strictly greater
            meta[7] = 0u;                      // atomic ctr: equal
        }
    }
}

// ---------------- Pass 4: histogram of key[15:0] restricted to hi bin ----------
__global__ void hist_lo_kernel(const float* __restrict__ obj, unsigned N,
                               const unsigned* __restrict__ meta,
                               unsigned* __restrict__ hist) {
    const unsigned hi = meta[0];
    unsigned i = blockIdx.x * blockDim.x + threadIdx.x;
    const unsigned stride = gridDim.x * blockDim.x;
    for (; i < N; i += stride) {
        __builtin_prefetch(obj + i + stride, 0, 1);
        unsigned k = fkey(obj[i]);
        if ((k >> 16) == hi) atomicAdd(&hist[k & 0xFFFFu], 1u);
    }
}

// ---------------- Pass 6: compact exact top-2000 (key,idx) pairs ---------------
__global__ void select_kernel(const float* __restrict__ obj, unsigned N,
                              unsigned* __restrict__ meta,
                              unsigned long long* __restrict__ sel) {
    const unsigned thrKey  = meta[3];
    const unsigned strictG = meta[4];
    const unsigned needEq  = meta[5];
    unsigned i = blockIdx.x * blockDim.x + threadIdx.x;
    const unsigned stride = gridDim.x * blockDim.x;
    for (; i < N; i += stride) {
        __builtin_prefetch(obj + i + stride, 0, 1);
        unsigned k = fkey(obj[i]);
        if (k >= thrKey) {
            unsigned long long v =
                ((unsigned long long)k << 32) | (unsigned)(~i); // desc key, asc idx
            if (k > thrKey) {
                unsigned p = atomicAdd(&meta[6], 1u);
                sel[p] = v;
            } else {
                unsigned p = atomicAdd(&meta[7], 1u);
                if (p < needEq) sel[strictG + p] = v;
            }
        }
    }
}

// ---------------- Pass 7: bitonic sort 2048 packed keys (descending) -----------
__global__ void sort_kernel(unsigned long long* __restrict__ sel) {
    __shared__ unsigned long long s[SORT_N];
    const unsigned t = threadIdx.x;              // 1024 threads
    s[t] = sel[t];
    s[t + 1024u] = sel[t + 1024u];
    __syncthreads();
    for (unsigned k = 2; k <= SORT_N; k <<= 1) {
        for (unsigned j = k >> 1; j > 0; j >>= 1) {
            for (unsigned i = t; i < SORT_N; i += 1024u) {
                unsigned ixj = i ^ j;
                if (ixj > i) {
                    bool desc = ((i & k) == 0u);
                    unsigned long long a = s[i], b = s[ixj];
                    if (desc ? (a < b) : (a > b)) { s[i] = b; s[ixj] = a; }
                }
            }
            __syncthreads();
        }
    }
    sel[t] = s[t];
    sel[t + 1024u] = s[t + 1024u];
}

// ---------------- Pass 8: gather + decode + clip + min-size -------------------
__global__ void decode_kernel(const unsigned long long* __restrict__ sel,
                              const float4* __restrict__ anchor,
                              const float4* __restrict__ delta,
                              const int* __restrict__ ph, const int* __restrict__ pw,
                              float4* __restrict__ boxes,
                              unsigned* __restrict__ valid) {
    unsigned i = blockIdx.x * blockDim.x + threadIdx.x;
    if (i >= PRE_NMS) return;
    unsigned idx = ~(unsigned)(sel[i] & 0xFFFFFFFFull);
    const float W = (float)(*pw), H = (float)(*ph);
    float4 a = anchor[idx];
    float4 d = delta[idx];
    float w  = a.z - a.x, h = a.w - a.y;
    float cx = a.x + 0.5f * w, cy = a.y + 0.5f * h;
    float dw = fminf(d.z, BBOX_CLIP), dh = fminf(d.w, BBOX_CLIP);
    float pcx = d.x * w + cx, pcy = d.y * h + cy;
    float pw_ = expf(dw) * w, ph_ = expf(dh) * h;
    float x1 = fminf(fmaxf(pcx - 0.5f * pw_, 0.f), W);
    float y1 = fminf(fmaxf(pcy - 0.5f * ph_, 0.f), H);
    float x2 = fminf(fmaxf(pcx + 0.5f * pw_, 0.f), W);
    float y2 = fminf(fmaxf(pcy + 0.5f * ph_, 0.f), H);
    boxes[i] = make_float4(x1, y1, x2, y2);
    valid[i] = ((x2 - x1) >= MIN_SIZE) && ((y2 - y1) >= MIN_SIZE);
}

// ---------------- Pass 9: IoU suppression bitmasks (32-thread = 1 wave) -------
__global__ void nms_mask_kernel(const float4* __restrict__ boxes,
                                unsigned* __restrict__ masks) {
    __shared__ float4 cb[32];
    const unsigned colBlk = blockIdx.x, rowBlk = blockIdx.y;
    const unsigned col0 = colBlk * 32u;
    const unsigned row  = rowBlk * 32u + threadIdx.x;
    unsigned c = col0 + threadIdx.x;
    cb[threadIdx.x] = (c < PRE_NMS) ? boxes[c] : make_float4(0.f, 0.f, 0.f, 0.f);
    __syncthreads();
    if (row >= PRE_NMS) return;
    float4 rb = boxes[row];
    float rArea = (rb.z - rb.x) * (rb.w - rb.y);
    unsigned m = 0u;
#pragma unroll 4
    for (unsigned q = 0; q < 32u; ++q) {
        unsigned col = col0 + q;
        if (col > row && col < PRE_NMS) {
            float4 b = cb[q];
            float xx1 = fmaxf(rb.x, b.x), yy1 = fmaxf(rb.y, b.y);
            float xx2 = fminf(rb.z, b.z), yy2 = fminf(rb.w, b.w);
            float inter = fmaxf(xx2 - xx1, 0.f) * fmaxf(yy2 - yy1, 0.f);
            float bArea = (b.z - b.x) * (b.w - b.y);
            float iou = inter / (rArea + bArea - inter + 1e-9f);
            if (iou > NMS_THR) m |= (1u << q);
        }
    }
    masks[row * MASK_WORDS + colBlk] = m;
}

// ---------------- Pass 10: greedy sequential scan (reference semantics) -------
__global__ void nms_scan_kernel(const unsigned* __restrict__ masks,
                                const unsigned* __restrict__ valid,
                                unsigned* __restrict__ keep) {
    __shared__ unsigned remv[MASK_WORDS];
    __shared__ int curKeep;
    const unsigned t = threadIdx.x;   // 64 threads
    for (unsigned w = t; w < MASK_WORDS; w += blockDim.x) remv[w] = 0u;
    __syncthreads();
    for (unsigned i = 0; i < PRE_NMS; ++i) {
        if (t == 0) {
            unsigned w = i >> 5, b = i & 31u;
            int k = (valid[i] != 0u) && !((remv[w] >> b) & 1u);
            keep[i] = (unsigned)k;
            curKeep = k;
        }
        __syncthreads();
        if (curKeep) {
            for (unsigned w = t; w < MASK_WORDS; w += blockDim.x)
                remv[w] |= masks[i * MASK_WORDS + w];
        }
        __syncthreads();
    }
}

// ---------------- Pass 11: ordered compaction into d_out ----------------------
__global__ void output_kernel(const float4* __restrict__ boxes,
                              const unsigned* __restrict__ keep,
                              float4* __restrict__ out) {
    __shared__ unsigned pos[PRE_NMS];
    const unsigned t = threadIdx.x, T = blockDim.x;
    for (unsigned i = t; i < POST_NMS; i += T) out[i] = make_float4(0.f, 0.f, 0.f, 0.f);
    if (t == 0) {
        unsigned c = 0;
        for (unsigned i = 0; i < PRE_NMS; ++i) { pos[i] = c; c += keep[i] ? 1u : 0u; }
    }
    __syncthreads();
    for (unsigned i = t; i < PRE_NMS; i += T)
        if (keep[i] && pos[i] < POST_NMS) out[pos[i]] = boxes[i];
}

// ---------------- launch ----------------
extern "C" void kernel_launch(void* const* d_in, const int* in_sizes, int n_in,
                              void* d_out, int out_size, void* d_ws, size_t ws_size,
                              hipStream_t stream) {
    (void)n_in; (void)out_size; (void)ws_size;
    const float4* anchor = (const float4*)d_in[0];
    const float*  obj    = (const float*)d_in[1];
    const float4* delta  = (const float4*)d_in[2];
    const int*    img_h  = (const int*)d_in[3];
    const int*    img_w  = (const int*)d_in[4];
    const unsigned N = (unsigned)in_sizes[1];

    // workspace layout (u32 units)
    unsigned* ws_u32 = (unsigned*)d_ws;
    unsigned*           hist  = ws_u32;                               // 65536
    unsigned*           meta  = ws_u32 + 65536;                       // 32
    unsigned long long* sel   = (unsigned long long*)(ws_u32 + 65568);// 2048 u64
    float4*             boxes = (float4*)(ws_u32 + 69664);            // 2048 f4
    unsigned*           valid = ws_u32 + 77856;                       // 2048
    unsigned*           keep  = ws_u32 + 79904;                       // 2048
    unsigned*           masks = ws_u32 + 81952;                       // 2000*63

    const unsigned zero1 = 69664;  // hist + meta + sel
    zero_u32_kernel<<<(zero1 + 255) / 256, 256, 0, stream>>>(ws_u32, zero1);
    // 512 blocks x 1024-elem tiles -> ~4 tiles/block: the 2-deep async
    // pipeline actually overlaps HBM fetch with LDS binning.
    hist_hi_kernel<<<512, 256, 0, stream>>>(obj, N, hist);
    scan_hist_kernel<<<1, 256, 0, stream>>>(hist, meta, 0);
    zero_u32_kernel<<<65536 / 256, 256, 0, stream>>>(hist, 65536);
    hist_lo_kernel<<<2048, 256, 0, stream>>>(obj, N, meta, hist);
    scan_hist_kernel<<<1, 256, 0, stream>>>(hist, meta, 1);
    select_kernel<<<2048, 256, 0, stream>>>(obj, N, meta, sel);
    sort_kernel<<<1, 1024, 0, stream>>>(sel);
    decode_kernel<<<(PRE_NMS + 255) / 256, 256, 0, stream>>>(sel, anchor, delta,
                                                             img_h, img_w, boxes, valid);
    nms_mask_kernel<<<dim3(MASK_WORDS, MASK_WORDS), 32, 0, stream>>>(boxes, masks);
    nms_scan_kernel<<<1, 64, 0, stream>>>(masks, valid, keep);
    output_kernel<<<1, 256, 0, stream>>>(boxes, keep, (float4*)d_out);
}